// AdaptiveLIF_4776003633771
// MI455X (gfx1250) — compile-verified
//
#include <hip/hip_runtime.h>

// Adaptive LIF scan: spikes[t,b,c,h,w] for
//   v = v*decay(b,c) + I[t];  s = (v > vth(b)) ? 1 : 0;  v = s ? 0 : v
// Shapes fixed by the reference: T=16, B=8, C=64, H=64, W=64 (float32).
//
// Memory-bound streaming kernel: 128 MB in + 128 MB out, ~11.5 us floor at
// 23.3 TB/s. One thread owns 4 consecutive w-elements -> b128 load/store per
// timestep, non-temporal hints (data touched exactly once, footprint > 192 MB
// L2), and gfx1250 global_prefetch_b8 for the t+1 slice (8 MB stride, always
// a cold miss otherwise).

typedef float v4f __attribute__((ext_vector_type(4)));

#define T_STEPS 16
#define B_DIM   8
#define C_DIM   64
#define HW_DIM  4096                         // 64*64
#define PLANE   (B_DIM * C_DIM * HW_DIM)     // elements per timestep = 2,097,152
#define VEC4_PER_PLANE (PLANE / 4)           // 524,288 float4 per timestep
#define VEC4_PER_BC    (HW_DIM / 4)          // 1024 float4 per (b,c) image

__global__ __launch_bounds__(256) void
adaptive_lif_scan_kernel(const float* __restrict__ I,
                         const float* __restrict__ tau,
                         const float* __restrict__ vth,
                         float* __restrict__ out)
{
    const int idx = blockIdx.x * blockDim.x + threadIdx.x;  // float4 index in one t-slice
    const int bc  = idx >> 10;        // / VEC4_PER_BC  -> b*C + c
    const int b   = bc >> 6;          // / C_DIM

    // Per-neuron constants (tiny, cache-resident; recompute per thread).
    const float tauv  = fmaxf(tau[bc], 0.001f);
    const float decay = expf(-1.0f / tauv);
    const float vthv  = fmaxf(vth[b], 0.001f);

    const v4f* __restrict__ pI = (const v4f*)I + idx;
    v4f* __restrict__       pO = (v4f*)out + idx;

    v4f v = (v4f)0.0f;

    #pragma unroll
    for (int t = 0; t < T_STEPS; ++t) {
        if (t + 1 < T_STEPS) {
            // gfx1250: global_prefetch_b8 — pull the t+1 slice (8 MB away,
            // guaranteed cold) toward the caches while the dependent scan
            // math for step t runs. Does not touch LOADcnt.
            __builtin_prefetch((const void*)(pI + (size_t)(t + 1) * VEC4_PER_PLANE), 0, 0);
        }

        // Streaming b128 load with TH=NT (touched exactly once).
        v4f x = __builtin_nontemporal_load(pI + (size_t)t * VEC4_PER_PLANE);

        v4f s;
        #pragma unroll
        for (int k = 0; k < 4; ++k) {
            float vv = fmaf(v[k], decay, x[k]);   // leak + integrate
            bool  sp = vv > vthv;                 // spike
            s[k] = sp ? 1.0f : 0.0f;
            v[k] = sp ? 0.0f : vv;                // hard reset
        }

        // Streaming b128 store with TH=NT (write-once, don't rinse L2).
        __builtin_nontemporal_store(s, pO + (size_t)t * VEC4_PER_PLANE);
    }
}

extern "C" void kernel_launch(void* const* d_in, const int* in_sizes, int n_in,
                              void* d_out, int out_size, void* d_ws, size_t ws_size,
                              hipStream_t stream)
{
    (void)in_sizes; (void)n_in; (void)out_size; (void)d_ws; (void)ws_size;

    const float* I   = (const float*)d_in[0];   // (T,B,C,H,W) f32
    const float* tau = (const float*)d_in[1];   // (B,C)       f32
    const float* vth = (const float*)d_in[2];   // (B,1)       f32
    float*       out = (float*)d_out;           // (T,B,C,H,W) f32

    const int threads = 256;                    // 8 wave32 per block
    const int blocks  = VEC4_PER_PLANE / threads;  // 2048 blocks

    adaptive_lif_scan_kernel<<<blocks, threads, 0, stream>>>(I, tau, vth, out);
}